// ManhattanRelPos3d_10960756539752
// MI455X (gfx1250) — compile-verified
//
#include <hip/hip_runtime.h>

typedef __attribute__((ext_vector_type(2))) float v2f;
typedef __attribute__((ext_vector_type(8))) float v8f;

#define GAMMA 0.9f
// H = W = 32, C = 8, P = 8192, out = P x P f32 (256 MB)

__global__ __launch_bounds__(256) void ManhattanRelPos3d_wmma(
    const float* __restrict__ decay,    // 8  (per-head == per-c1)
    const float* __restrict__ spec_pe,  // 64 [c1*8 + c2]
    float* __restrict__ out)            // 8192*8192
{
  __shared__ float s_pw[64];    // gamma^d
  __shared__ float s_g32[1024]; // gamma^|i-j| for 32x32 (serves gh and gw)
  __shared__ float s_Sx[32];    // row sums of s_g32
  __shared__ float s_A[64];     // A[c1*8+c2]
  __shared__ float s_B[64];     // B[c1*8+c2]

  const int tid = threadIdx.x;

  if (tid == 0) {
    float p = 1.0f;
    for (int d = 0; d < 64; ++d) { s_pw[d] = p; p *= GAMMA; }
  }
  __syncthreads();

  for (int idx = tid; idx < 1024; idx += 256) {
    int i = idx >> 5, j = idx & 31;
    int d = i - j; d = d < 0 ? -d : d;
    s_g32[idx] = s_pw[d];
  }
  if (tid < 32) {
    float s = 0.0f;
    for (int j = 0; j < 32; ++j) { int d = tid - j; d = d < 0 ? -d : d; s += s_pw[d]; }
    s_Sx[tid] = s;
  }
  if (tid < 64) {
    int c1 = tid >> 3, c2 = tid & 7;
    int d = c1 - c2; d = d < 0 ? -d : d;
    float A = s_pw[d] * decay[c1];
    float Sc = 0.0f;
    for (int k = 0; k < 8; ++k) { int dd = c1 - k; dd = dd < 0 ? -dd : dd; Sc += s_pw[dd]; }
    s_A[tid] = A;
    s_B[tid] = A * spec_pe[tid] * Sc;
  }
  __syncthreads();

  const int strip = blockIdx.x;   // 0..511
  const int h1  = strip >> 4;     // 0..31
  const int w1p = strip & 15;     // w1 = 2*w1p + a

  const int wave = tid >> 5;
  const int lane = tid & 31;
  const int lj   = lane & 15;           // M (A-op) / N (B-op/D) index
  const int hiK  = (lane >> 4) << 1;    // K base: 0 (lanes 0-15) or 2 (lanes 16-31)
  const int aA   = lj >> 3;             // 'a' of row i = (a, c1)
  const int c1A  = lj & 7;

  // ---- constant B operands: bn1[kk] = slice kk of I2 (x) A, bn2 of I2 (x) B ----
  v2f bn1[4], bn2[4];
#pragma unroll
  for (int kk = 0; kk < 4; ++kk) {
#pragma unroll
    for (int v = 0; v < 2; ++v) {
      int k  = 4 * kk + v + hiK;          // global K 0..15 = (b', c)
      int bp = k >> 3;
      int c  = k & 7;
      bool on = (bp == (lj >> 3));        // matches b(j)
      bn1[kk][v] = on ? s_A[c * 8 + (lj & 7)] : 0.0f;
      bn2[kk][v] = on ? s_B[c * 8 + (lj & 7)] : 0.0f;
    }
  }

  // ---- A-operand sparsity masks: [c1(i) == c(k)] ----
  float me[2], mo[2];
#pragma unroll
  for (int v = 0; v < 2; ++v) {
    me[v] = (c1A == (v + hiK))     ? 1.0f : 0.0f;  // kk even: c = v + hiK
    mo[v] = (c1A == (v + hiK + 4)) ? 1.0f : 0.0f;  // kk odd : c = v + hiK + 4
  }

  const float T_a   = s_Sx[h1] * s_Sx[2 * w1p + aA] * (1.0f / 8192.0f);
  const int   rowoff = (2 * w1p + aA) * 32;        // gw row for this lane's 'a'

  // store-side p base: lanes 0-15 -> a=0 rows (M=v), lanes 16-31 -> a=1 (M=v+8)
  const int  aS    = lane >> 4;
  const long pbase = (long)((h1 * 32 + 2 * w1p + aS) * 8);

  for (int h2 = wave; h2 < 32; h2 += 8) {
    const float ghh = s_g32[h1 * 32 + h2];
    float* optr = out + pbase * 8192 + (long)h2 * 256 + lj;

    for (int w2p = 0; w2p < 16; ++w2p) {
      float gw0 = s_g32[rowoff + 2 * w2p + 0];
      float gw1 = s_g32[rowoff + 2 * w2p + 1];
      float P0 = ghh * gw0, P1 = ghh * gw1;
      float Q0 = P0 * T_a,  Q1 = P1 * T_a;

      v2f a0, a1, a2, a3, a4, a5, a6, a7;
      a0[0] = me[0] * P0; a0[1] = me[1] * P0;   // kk0: b'=0, c 0-3
      a1[0] = mo[0] * P0; a1[1] = mo[1] * P0;   // kk1: b'=0, c 4-7
      a2[0] = me[0] * P1; a2[1] = me[1] * P1;   // kk2: b'=1, c 0-3
      a3[0] = mo[0] * P1; a3[1] = mo[1] * P1;   // kk3: b'=1, c 4-7
      a4[0] = me[0] * Q0; a4[1] = me[1] * Q0;
      a5[0] = mo[0] * Q0; a5[1] = mo[1] * Q0;
      a6[0] = me[0] * Q1; a6[1] = me[1] * Q1;
      a7[0] = mo[0] * Q1; a7[1] = mo[1] * Q1;

      v8f acc = {};
      acc = __builtin_amdgcn_wmma_f32_16x16x4_f32(false, a0, false, bn1[0], (short)0, acc, false, false);
      acc = __builtin_amdgcn_wmma_f32_16x16x4_f32(false, a1, false, bn1[1], (short)0, acc, false, false);
      acc = __builtin_amdgcn_wmma_f32_16x16x4_f32(false, a2, false, bn1[2], (short)0, acc, false, false);
      acc = __builtin_amdgcn_wmma_f32_16x16x4_f32(false, a3, false, bn1[3], (short)0, acc, false, false);
      acc = __builtin_amdgcn_wmma_f32_16x16x4_f32(false, a4, false, bn2[0], (short)0, acc, false, false);
      acc = __builtin_amdgcn_wmma_f32_16x16x4_f32(false, a5, false, bn2[1], (short)0, acc, false, false);
      acc = __builtin_amdgcn_wmma_f32_16x16x4_f32(false, a6, false, bn2[2], (short)0, acc, false, false);
      acc = __builtin_amdgcn_wmma_f32_16x16x4_f32(false, a7, false, bn2[3], (short)0, acc, false, false);

      float* o = optr + w2p * 16;
#pragma unroll
      for (int v = 0; v < 8; ++v)
        __builtin_nontemporal_store(acc[v], o + (long)v * 8192);
    }
  }
}

extern "C" void kernel_launch(void* const* d_in, const int* in_sizes, int n_in,
                              void* d_out, int out_size, void* d_ws, size_t ws_size,
                              hipStream_t stream) {
  // d_in[0] = x (unused by the math), d_in[1] = decay (8), d_in[2] = spec_pe (64)
  const float* decay   = (const float*)d_in[1];
  const float* spec_pe = (const float*)d_in[2];
  float* out = (float*)d_out;  // 8192*8192 f32

  ManhattanRelPos3d_wmma<<<dim3(512), dim3(256), 0, stream>>>(decay, spec_pe, out);
}